// PSDLoss_83081847373853
// MI455X (gfx1250) — compile-verified
//
#include <hip/hip_runtime.h>
#include <math.h>

typedef float v2f __attribute__((ext_vector_type(2)));
typedef float v4f __attribute__((ext_vector_type(4)));
typedef float v8f __attribute__((ext_vector_type(8)));
typedef int   v4i __attribute__((ext_vector_type(4)));

#define HW    512
#define HW2   (512 * 512)
#define NIMG  24
#define NBINS 362
#define EPSF  1e-8f
#define APAD  36   // LDS row stride (floats) for the staged A panel: bank-conflict-free

#if defined(__has_builtin)
#  if __has_builtin(__builtin_amdgcn_global_load_async_to_lds_b128) && \
      __has_builtin(__builtin_amdgcn_s_wait_asynccnt)
#    define USE_ASYNC_LDS 1
#  endif
#endif
#ifndef USE_ASYNC_LDS
#  define USE_ASYNC_LDS 0
#endif

#define AS1 __attribute__((address_space(1)))
#define AS3 __attribute__((address_space(3)))

// radius bin of unshifted freq position (u,v), matching fftshift-ed reference grid
__device__ __forceinline__ int radial_bin(int u, int v) {
    int i = (u + 256) & 511;
    int j = (v + 256) & 511;
    float dy = (float)i - 255.5f;
    float dx = (float)j - 255.5f;
    return (int)sqrtf(dx * dx + dy * dy);   // floor, r >= 0
}

// DFT matrix F[k][n] = exp(-2*pi*i*k*n/512); exactly symmetric via (k*n) mod 512.
__global__ void k_build_F(float* __restrict__ Fre, float* __restrict__ Fim) {
    int idx = blockIdx.x * blockDim.x + threadIdx.x;
    if (idx >= HW2) return;
    int k = idx >> 9, n = idx & 511;
    int t = (k * n) & 511;
    float ang = -6.283185307179586f * (float)t * (1.0f / 512.0f);
    float s, c;
    sincosf(ang, &s, &c);
    Fre[idx] = c;
    Fim[idx] = s;
}

__global__ void k_zero(float* __restrict__ p, int n, float* __restrict__ out) {
    int i = blockIdx.x * blockDim.x + threadIdx.x;
    if (i < n) p[i] = 0.0f;
    if (i == 0) *out = 0.0f;
}

// counts[b] = #positions with r_int == b (float, matches reference COUNTS)
__global__ void k_counts(float* __restrict__ counts) {
    __shared__ float sb[NBINS];
    for (int i = threadIdx.x; i < NBINS; i += blockDim.x) sb[i] = 0.0f;
    __syncthreads();
    int idx = blockIdx.x * blockDim.x + threadIdx.x;
    int u = idx >> 9, v = idx & 511;
    atomicAdd(&sb[radial_bin(u, v)], 1.0f);
    __syncthreads();
    for (int i = threadIdx.x; i < NBINS; i += blockDim.x)
        if (sb[i] != 0.0f) atomicAdd(&counts[i], sb[i]);
}

// Stage 1: T1 = X (real) * F (complex), row-FFT. F symmetric -> B fragments are
// contiguous v2f row loads. Output stored TRANSPOSED: T1s[k2][u] so stage 2's
// B fragments are contiguous; each lane owns 8 contiguous floats -> b128 stores.
__global__ void __launch_bounds__(128)
k_gemm1(const float* __restrict__ X,
        const float* __restrict__ Fre, const float* __restrict__ Fim,
        float* __restrict__ T1re, float* __restrict__ T1im) {
    const int lane = threadIdx.x & 31;
    const int wave = threadIdx.x >> 5;
    const int tile = blockIdx.x * 4 + wave;          // 1024 tiles (32x32)
    const int tm = tile >> 5, tn = tile & 31;
    const int h = lane >> 4;
    const int m16 = lane & 15;
    const int arow = (tm << 4) + m16;                // A row (image row)
    const int ncol = (tn << 4) + m16;                // output column k2

    v8f accR = {};
    v8f accI = {};
#pragma unroll 4
    for (int k0 = 0; k0 < 512; k0 += 4) {
        const int kb = k0 + 2 * h;
        v2f a  = *(const v2f*)(X   + arow * 512 + kb);
        v2f br = *(const v2f*)(Fre + ncol * 512 + kb);   // F[kb][ncol] via symmetry
        v2f bi = *(const v2f*)(Fim + ncol * 512 + kb);
        accR = __builtin_amdgcn_wmma_f32_16x16x4_f32(false, a, false, br, (short)0, accR, false, false);
        accI = __builtin_amdgcn_wmma_f32_16x16x4_f32(false, a, false, bi, (short)0, accI, false, false);
    }
    // lane holds T1[u][ncol] for u = tm*16 + 8h + r (r=0..7, contiguous)
    const int base = ncol * 512 + (tm << 4) + (h << 3);
    *(v4f*)(T1re + base)     = __builtin_shufflevector(accR, accR, 0, 1, 2, 3);
    *(v4f*)(T1re + base + 4) = __builtin_shufflevector(accR, accR, 4, 5, 6, 7);
    *(v4f*)(T1im + base)     = __builtin_shufflevector(accI, accI, 0, 1, 2, 3);
    *(v4f*)(T1im + base + 4) = __builtin_shufflevector(accI, accI, 4, 5, 6, 7);
}

// Stage 2: T2 = F (complex) * T1 (complex), column-FFT, fused epilogue.
// All 4 waves of a block share tm -> the 16-row A panel of F is staged in LDS
// via async-to-LDS b128 (double buffered, ASYNCcnt protocol) when available.
// isPred=1: PF <- |P|^2 ; bins[r_int] += 20*ln|P+eps| (LDS-reduced)
// isPred=0: PF <- (PF - |T|^2)^2  in place
__global__ void __launch_bounds__(128)
k_gemm2(const float* __restrict__ Fre, const float* __restrict__ Fim,
        const float* __restrict__ T1re, const float* __restrict__ T1im,
        float* __restrict__ PF, float* __restrict__ bins, int isPred) {
    __shared__ float sb[NBINS];
    if (isPred) {
        for (int i = threadIdx.x; i < NBINS; i += blockDim.x) sb[i] = 0.0f;
        __syncthreads();
    }
    const int lane = threadIdx.x & 31;
    const int wave = threadIdx.x >> 5;
    const int tile = blockIdx.x * 4 + wave;
    const int tm = tile >> 5, tn = tile & 31;        // tm identical across the block
    const int h = lane >> 4;
    const int m16 = lane & 15;
    const int ncol = (tn << 4) + m16;

    v8f accR = {};
    v8f accI = {};

#if USE_ASYNC_LDS
    __shared__ float sAre[2][16][APAD];
    __shared__ float sAim[2][16][APAD];
    const int t = threadIdx.x;
    const int lrow = t >> 3;                 // 0..15 : panel row
    const int lcol = (t & 7) << 2;           // 0,4,...,28 : 4 floats per thread
    const float* gRe = Fre + ((tm << 4) + lrow) * 512 + lcol;
    const float* gIm = Fim + ((tm << 4) + lrow) * 512 + lcol;

    // preload chunk 0
    __builtin_amdgcn_global_load_async_to_lds_b128((AS1 v4i*)(gRe),
                                                   (AS3 v4i*)&sAre[0][lrow][lcol], 0, 0);
    __builtin_amdgcn_global_load_async_to_lds_b128((AS1 v4i*)(gIm),
                                                   (AS3 v4i*)&sAim[0][lrow][lcol], 0, 0);
    for (int c = 0; c < 16; ++c) {           // 16 chunks of K=32
        const int buf = c & 1;
        if (c < 15) {
            __builtin_amdgcn_global_load_async_to_lds_b128((AS1 v4i*)(gRe + (c + 1) * 32),
                                                           (AS3 v4i*)&sAre[buf ^ 1][lrow][lcol], 0, 0);
            __builtin_amdgcn_global_load_async_to_lds_b128((AS1 v4i*)(gIm + (c + 1) * 32),
                                                           (AS3 v4i*)&sAim[buf ^ 1][lrow][lcol], 0, 0);
            __builtin_amdgcn_s_wait_asynccnt(2);   // chunk c complete (in-order)
        } else {
            __builtin_amdgcn_s_wait_asynccnt(0);
        }
        __syncthreads();                      // cross-wave LDS visibility
#pragma unroll
        for (int s = 0; s < 8; ++s) {
            const int kl = 4 * s + 2 * h;
            const int kb = 32 * c + kl;
            v2f ar  = *(const v2f*)&sAre[buf][m16][kl];
            v2f ai  = *(const v2f*)&sAim[buf][m16][kl];
            v2f ain = -ai;
            v2f br = *(const v2f*)(T1re + ncol * 512 + kb);
            v2f bi = *(const v2f*)(T1im + ncol * 512 + kb);
            accR = __builtin_amdgcn_wmma_f32_16x16x4_f32(false, ar,  false, br, (short)0, accR, false, false);
            accR = __builtin_amdgcn_wmma_f32_16x16x4_f32(false, ain, false, bi, (short)0, accR, false, false);
            accI = __builtin_amdgcn_wmma_f32_16x16x4_f32(false, ar,  false, bi, (short)0, accI, false, false);
            accI = __builtin_amdgcn_wmma_f32_16x16x4_f32(false, ai,  false, br, (short)0, accI, false, false);
        }
        __syncthreads();                      // protect buffer before re-issue
    }
#else
    const int arow = (tm << 4) + m16;
#pragma unroll 2
    for (int k0 = 0; k0 < 512; k0 += 4) {
        const int kb = k0 + 2 * h;
        v2f ar  = *(const v2f*)(Fre + arow * 512 + kb);
        v2f ai  = *(const v2f*)(Fim + arow * 512 + kb);
        v2f ain = -ai;
        v2f br = *(const v2f*)(T1re + ncol * 512 + kb);
        v2f bi = *(const v2f*)(T1im + ncol * 512 + kb);
        accR = __builtin_amdgcn_wmma_f32_16x16x4_f32(false, ar,  false, br, (short)0, accR, false, false);
        accR = __builtin_amdgcn_wmma_f32_16x16x4_f32(false, ain, false, bi, (short)0, accR, false, false);
        accI = __builtin_amdgcn_wmma_f32_16x16x4_f32(false, ar,  false, bi, (short)0, accI, false, false);
        accI = __builtin_amdgcn_wmma_f32_16x16x4_f32(false, ai,  false, br, (short)0, accI, false, false);
    }
#endif

#pragma unroll
    for (int r = 0; r < 8; ++r) {
        int u = (tm << 4) + r + 8 * h;
        int v = ncol;
        float re = accR[r], im = accI[r];
        int idx = u * 512 + v;
        if (isPred) {
            PF[idx] = re * re + im * im;
            float er = re + EPSF;
            float mag = 10.0f * logf(er * er + im * im);   // 20*ln|P+eps|
            atomicAdd(&sb[radial_bin(u, v)], mag);
        } else {
            float t2 = re * re + im * im;
            float d = PF[idx] - t2;
            PF[idx] = d * d;
        }
    }
    if (isPred) {
        __syncthreads();
        for (int i = threadIdx.x; i < NBINS; i += blockDim.x)
            if (sb[i] != 0.0f) atomicAdd(&bins[i], sb[i]);
    }
}

// Per (n,c): binmean = bins/counts; psd = binmean[1:361]; min/max normalize;
// psd_ext = [psd[0], psd[0..359], 0]  (length 362)
__global__ void __launch_bounds__(128)
k_psd(const float* __restrict__ bins, const float* __restrict__ counts,
      float* __restrict__ psde) {
    const int img = blockIdx.x;
    const int tid = threadIdx.x;
    __shared__ float sm[NBINS];
    __shared__ float rmin[128], rmax[128];
    for (int b = tid; b < NBINS; b += 128) sm[b] = bins[img * NBINS + b] / counts[b];
    __syncthreads();
    float lmin = 3.0e38f, lmax = -3.0e38f;
    for (int j = tid; j < 360; j += 128) {
        float p = sm[j + 1];
        lmin = fminf(lmin, p);
        lmax = fmaxf(lmax, p);
    }
    rmin[tid] = lmin;
    rmax[tid] = lmax;
    __syncthreads();
    for (int s = 64; s > 0; s >>= 1) {
        if (tid < s) {
            rmin[tid] = fminf(rmin[tid], rmin[tid + s]);
            rmax[tid] = fmaxf(rmax[tid], rmax[tid + s]);
        }
        __syncthreads();
    }
    const float pmin = rmin[0];
    const float inv = 1.0f / (rmax[0] - pmin);
    for (int k2 = tid; k2 < NBINS; k2 += 128) {
        float val;
        if (k2 == 361) {
            val = 0.0f;                                   // appended zero (post-normalize)
        } else {
            int j = (k2 == 0) ? 0 : (k2 - 1);
            val = (sm[j + 1] - pmin) * inv;
        }
        psde[img * NBINS + k2] = val;
    }
}

// loss = mean over 24*512*512 of clip(1 - psd_ext[r_int], 0, 1) * fd   (NaN -> 0)
__global__ void __launch_bounds__(256)
k_loss(const float* __restrict__ PF, const float* __restrict__ psde,
       float* __restrict__ out) {
    __shared__ float red[256];
    const long long N = (long long)NIMG * HW2;
    float acc = 0.0f;
    for (long long idx = (long long)blockIdx.x * 256 + threadIdx.x; idx < N;
         idx += (long long)gridDim.x * 256) {
        int img = (int)(idx >> 18);                      // HW2 == 2^18
        int pos = (int)(idx & (HW2 - 1));
        int u = pos >> 9, v = pos & 511;
        float m = psde[img * NBINS + radial_bin(u, v)];
        float w = 1.0f - m;
        if (!(w == w)) w = 0.0f;                         // NaN -> 0
        w = fminf(fmaxf(w, 0.0f), 1.0f);
        acc += w * PF[idx];
    }
    red[threadIdx.x] = acc;
    __syncthreads();
    for (int s = 128; s > 0; s >>= 1) {
        if (threadIdx.x < s) red[threadIdx.x] += red[threadIdx.x + s];
        __syncthreads();
    }
    if (threadIdx.x == 0) atomicAdd(out, red[0] * (1.0f / 6291456.0f));
}

extern "C" void kernel_launch(void* const* d_in, const int* in_sizes, int n_in,
                              void* d_out, int out_size, void* d_ws, size_t ws_size,
                              hipStream_t stream) {
    (void)in_sizes; (void)n_in; (void)out_size; (void)ws_size;
    const float* pred = (const float*)d_in[0];
    const float* tgt  = (const float*)d_in[1];
    float* out = (float*)d_out;

    float* ws     = (float*)d_ws;                 // ~29.5 MB total
    float* Fre    = ws;                           // 512*512
    float* Fim    = Fre + HW2;                    // 512*512
    float* T1re   = Fim + HW2;                    // 512*512 transposed, reused per image
    float* T1im   = T1re + HW2;                   // 512*512
    float* PF     = T1im + HW2;                   // 24 * 512*512  (|P|^2 then fd)
    float* bins   = PF + (size_t)NIMG * HW2;      // 24 * 362
    float* counts = bins + NIMG * NBINS;          // 362
    float* psde   = counts + NBINS;               // 24 * 362

    k_build_F<<<HW2 / 256, 256, 0, stream>>>(Fre, Fim);
    const int zn = NIMG * NBINS + NBINS + NIMG * NBINS;   // bins+counts+psde contiguous
    k_zero<<<(zn + 255) / 256, 256, 0, stream>>>(bins, zn, out);
    k_counts<<<HW2 / 256, 256, 0, stream>>>(counts);

    // Pred: FFT + |P|^2 + radial mag bins
    for (int img = 0; img < NIMG; ++img) {
        k_gemm1<<<256, 128, 0, stream>>>(pred + (size_t)img * HW2, Fre, Fim, T1re, T1im);
        k_gemm2<<<256, 128, 0, stream>>>(Fre, Fim, T1re, T1im,
                                         PF + (size_t)img * HW2, bins + img * NBINS, 1);
    }
    // Target: FFT + fd = (|P|^2 - |T|^2)^2 in place
    for (int img = 0; img < NIMG; ++img) {
        k_gemm1<<<256, 128, 0, stream>>>(tgt + (size_t)img * HW2, Fre, Fim, T1re, T1im);
        k_gemm2<<<256, 128, 0, stream>>>(Fre, Fim, T1re, T1im,
                                         PF + (size_t)img * HW2, nullptr, 0);
    }

    k_psd<<<NIMG, 128, 0, stream>>>(bins, counts, psde);
    k_loss<<<512, 256, 0, stream>>>(PF, psde, out);
}